// MyFlatten_5454608466609
// MI455X (gfx1250) — compile-verified
//
#include <hip/hip_runtime.h>

typedef __attribute__((ext_vector_type(2))) float v2f;
typedef __attribute__((ext_vector_type(8))) float v8f;

#define NBATCH 16
#define BLOCK 256
#define EPS 1e-6f

__global__ void zero_out_kernel(float* __restrict__ out, int n) {
    int i = blockIdx.x * blockDim.x + threadIdx.x;
    if (i < n) out[i] = 0.0f;
}

__device__ __forceinline__ float wave_reduce_add(float v) {
    v += __shfl_xor(v, 1, 32);
    v += __shfl_xor(v, 2, 32);
    v += __shfl_xor(v, 4, 32);
    v += __shfl_xor(v, 8, 32);
    v += __shfl_xor(v, 16, 32);
    return v;
}

// One thread per edge; fully unrolled loop over 16 batches so ~64 independent
// 12-byte gathers are in flight per thread (L2-resident vertex tensor).
// Block reduction: wave32 shuffle tree -> LDS 16x8 partial matrix ->
// two chained V_WMMA_F32_16X16X4_F32 with all-ones B (exact f32 row sums).
__global__ __launch_bounds__(BLOCK)
void tet_cosloss_kernel(const float* __restrict__ vertices,
                        const int* __restrict__ v0s, const int* __restrict__ v1s,
                        const int* __restrict__ v2s, const int* __restrict__ v3s,
                        float* __restrict__ block_out,
                        int ne, int nv, int use_atomic)
{
    __shared__ float lds_red[8 * NBATCH];   // [wave][batch]

    const int e = blockIdx.x * BLOCK + threadIdx.x;

    float acc[NBATCH];
#pragma unroll
    for (int b = 0; b < NBATCH; ++b) acc[b] = 0.0f;

    if (e < ne) {
        const int i0 = v0s[e];
        const int i1 = v1s[e];
        const int i2 = v2s[e];
        const int i3 = v3s[e];
        const size_t o0 = (size_t)i0 * 3;
        const size_t o1 = (size_t)i1 * 3;
        const size_t o2 = (size_t)i2 * 3;
        const size_t o3 = (size_t)i3 * 3;

#pragma unroll
        for (int b = 0; b < NBATCH; ++b) {
            const float* vb = vertices + (size_t)b * (size_t)nv * 3;

            const float p0x = vb[o0 + 0], p0y = vb[o0 + 1], p0z = vb[o0 + 2];
            const float p1x = vb[o1 + 0], p1y = vb[o1 + 1], p1z = vb[o1 + 2];
            const float p2x = vb[o2 + 0], p2y = vb[o2 + 1], p2z = vb[o2 + 2];
            const float p3x = vb[o3 + 0], p3y = vb[o3 + 1], p3z = vb[o3 + 2];

            const float ax = p1x - p0x, ay = p1y - p0y, az = p1z - p0z;
            const float ux = p2x - p0x, uy = p2y - p0y, uz = p2z - p0z;   // b for perp #1
            const float wx = p3x - p0x, wy = p3y - p0y, wz = p3z - p0z;   // b for perp #2

            const float al2 = ax * ax + ay * ay + az * az;
            const float ul2 = ux * ux + uy * uy + uz * uz;
            const float wl2 = wx * wx + wy * wy + wz * wz;

            const float al1 = sqrtf(al2 + EPS);
            const float ul1 = sqrtf(ul2 + EPS);
            const float wl1 = sqrtf(wl2 + EPS);

            const float au = ax * ux + ay * uy + az * uz;
            const float aw = ax * wx + ay * wy + az * wz;

            const float cos1 = au / (al1 * ul1 + EPS);
            const float cos2 = aw / (al1 * wl1 + EPS);
            const float sin1 = sqrtf(1.0f - cos1 * cos1 + EPS);
            const float sin2 = sqrtf(1.0f - cos2 * cos2 + EPS);

            const float s1 = au / (al2 + EPS);
            const float s2 = aw / (al2 + EPS);

            const float c1x = ux - ax * s1, c1y = uy - ay * s1, c1z = uz - az * s1;
            const float c2x = wx - ax * s2, c2y = wy - ay * s2, c2z = wz - az * s2;

            const float dotc  = c1x * c2x + c1y * c2y + c1z * c2z;
            const float denom = (ul1 * sin1) * (wl1 * sin2) + EPS;

            const float t = dotc / denom + 1.0f;
            acc[b] = t * t;
        }
    }

    // Wave32 shuffle reduction for each batch
    const int lane = threadIdx.x & 31;
    const int wave = threadIdx.x >> 5;
#pragma unroll
    for (int b = 0; b < NBATCH; ++b) acc[b] = wave_reduce_add(acc[b]);

    if (lane == 0) {
#pragma unroll
        for (int b = 0; b < NBATCH; ++b) lds_red[wave * NBATCH + b] = acc[b];
    }
    __syncthreads();

    // WMMA block reduction. All waves execute it (EXEC must be all-1s for
    // WMMA; blocks are always full). A[m][k] layout for f32 16x16x4:
    // m = lane%16, k = 2*(lane/16) + vgpr_index.
    {
        const int half = lane >> 4;     // 0 or 1
        const int m    = lane & 15;     // batch row

        v2f a1, a2, bones;
        a1.x = lds_red[(2 * half + 0) * NBATCH + m];
        a1.y = lds_red[(2 * half + 1) * NBATCH + m];
        a2.x = lds_red[(2 * half + 4) * NBATCH + m];
        a2.y = lds_red[(2 * half + 5) * NBATCH + m];
        bones.x = 1.0f;
        bones.y = 1.0f;

        v8f c = {};
        c = __builtin_amdgcn_wmma_f32_16x16x4_f32(false, a1, false, bones,
                                                  (short)0, c, false, false);
        c = __builtin_amdgcn_wmma_f32_16x16x4_f32(false, a2, false, bones,
                                                  (short)0, c, false, false);

        // D VGPR r: lanes 0-15 -> M=r (N=lane); lanes 16-31 -> M=r+8.
        // Take column N=0: lane 0 holds batches 0..7, lane 16 holds 8..15.
        if (wave == 0 && (lane == 0 || lane == 16)) {
            const int mbase = (lane == 16) ? 8 : 0;
            if (use_atomic) {
#pragma unroll
                for (int r = 0; r < 8; ++r)
                    atomicAdd(&block_out[mbase + r], c[r]);
            } else {
                float* dst = block_out + (size_t)blockIdx.x * NBATCH + mbase;
#pragma unroll
                for (int r = 0; r < 8; ++r) dst[r] = c[r];
            }
        }
    }
}

// Deterministic second-stage reduction: one block per batch.
__global__ __launch_bounds__(BLOCK)
void reduce_partials_kernel(const float* __restrict__ partials,
                            float* __restrict__ out, int nblocks)
{
    __shared__ float sdata[8];
    const int b = blockIdx.x;           // batch 0..15
    float s = 0.0f;
    for (int i = threadIdx.x; i < nblocks; i += BLOCK)
        s += partials[(size_t)i * NBATCH + b];
    s = wave_reduce_add(s);
    const int lane = threadIdx.x & 31;
    const int wave = threadIdx.x >> 5;
    if (lane == 0) sdata[wave] = s;
    __syncthreads();
    if (threadIdx.x == 0) {
        float t = 0.0f;
#pragma unroll
        for (int w = 0; w < 8; ++w) t += sdata[w];
        out[b] = t;
    }
}

extern "C" void kernel_launch(void* const* d_in, const int* in_sizes, int n_in,
                              void* d_out, int out_size, void* d_ws, size_t ws_size,
                              hipStream_t stream) {
    const float* vertices = (const float*)d_in[0];
    const int*   v0s      = (const int*)d_in[1];
    const int*   v1s      = (const int*)d_in[2];
    const int*   v2s      = (const int*)d_in[3];
    const int*   v3s      = (const int*)d_in[4];
    float*       out      = (float*)d_out;

    const int ne = in_sizes[1];                      // 300000 edges
    const int nb = out_size;                         // 16 batches
    const int nv = in_sizes[0] / (nb * 3);           // 100000 vertices

    const int nblocks = (ne + BLOCK - 1) / BLOCK;
    const size_t need = (size_t)nblocks * NBATCH * sizeof(float);

    if (ws_size >= need) {
        // Deterministic two-pass reduction through workspace.
        tet_cosloss_kernel<<<nblocks, BLOCK, 0, stream>>>(
            vertices, v0s, v1s, v2s, v3s, (float*)d_ws, ne, nv, /*use_atomic=*/0);
        reduce_partials_kernel<<<NBATCH, BLOCK, 0, stream>>>(
            (const float*)d_ws, out, nblocks);
    } else {
        // Fallback: zero output then atomic accumulation.
        zero_out_kernel<<<1, 32, 0, stream>>>(out, NBATCH);
        tet_cosloss_kernel<<<nblocks, BLOCK, 0, stream>>>(
            vertices, v0s, v1s, v2s, v3s, out, ne, nv, /*use_atomic=*/1);
    }
}